// ResamplerLayer_2534030704699
// MI455X (gfx1250) — compile-verified
//
#include <hip/hip_runtime.h>

// Trilinear resample, REPLICATE boundary, shapes hardcoded from reference:
//   inputs:        (B=2, D=128, H=128, W=128, C=8) f32
//   sample_coords: (B=2, 96, 96, 96, 3) f32
//   output:        (B=2, 96, 96, 96, 8) f32
//
// Memory-bound gather. MI455X strategy (23.3 TB/s HBM, 192 MB L2):
//  - input volume (134 MB) kept L2-resident: regular-temporal gather loads
//  - coords + output streamed with NT cache policy (no L2 pollution)
//  - coords staged memory->LDS with GLOBAL_LOAD_ASYNC_TO_LDS_B128 (ASYNCcnt,
//    no VGPR round-trip), s_wait_asynccnt + workgroup barrier, then
//    conflict-free LDS readback (bank = 3t mod 64 bijective per wave32)
//  - all 16 corner b128 loads issued as one clause per point (latency hiding)
//  - wave32: 256-thread blocks = 8 waves

typedef float __attribute__((ext_vector_type(4))) f4;

#define DIN   128
#define NPTS  (2 * 96 * 96 * 96)          // 1,769,472 (== 6912 * 256 exactly)
#define PER_BATCH (96 * 96 * 96)          // 884,736
#define THREADS 256
#define COORD_BYTES (THREADS * 3 * 4)     // 3072 B of dynamic LDS per block

__global__ __launch_bounds__(THREADS, 4)
void resample_trilinear_kernel(const float* __restrict__ in,
                               const float* __restrict__ coords,
                               float* __restrict__ out)
{
    // Dynamic LDS => starts at LDS offset 0 (no static __shared__ in kernel),
    // so the raw byte addresses used by the async DMA below are exact.
    extern __shared__ float scoord[];       // 3 KB

    const int t = threadIdx.x;
    const int blockStart = blockIdx.x * THREADS;

    // ---- Async-stage this block's 256 coordinate triplets (3072 B) ----
    // 192 lanes each DMA one 16 B chunk memory->LDS (coalesced, NT policy).
    // GVS mode: uniform SGPR base + per-lane 32-bit VGPR offset.
    const unsigned long long cbase =
        (unsigned long long)(const void*)(coords + (size_t)blockStart * 3);
    if (t < COORD_BYTES / 16) {
        unsigned lds_byte = (unsigned)(t * 16);   // LDS dest address
        unsigned g_off    = (unsigned)(t * 16);   // byte offset from cbase
        asm volatile(
            "global_load_async_to_lds_b128 %0, %1, %2 th:TH_LOAD_NT"
            :: "v"(lds_byte), "v"(g_off), "s"(cbase)
            : "memory");
    }
    // Wait for this wave's async DMAs to land in LDS, then block barrier.
    asm volatile("s_wait_asynccnt 0x0" ::: "memory");
    __syncthreads();

    const int p = blockStart + t;           // global point id, < 2^21
    const float cz = scoord[3 * t + 0];     // dim 0 -> D
    const float cy = scoord[3 * t + 1];     // dim 1 -> H
    const float cx = scoord[3 * t + 2];     // dim 2 -> W

    // ---- Trilinear setup (matches reference: floor, frac, clamp both corners) ----
    const float fz = floorf(cz), fy = floorf(cy), fx = floorf(cx);
    const float wz = cz - fz, wy = cy - fy, wx = cx - fx;

    const int iz = (int)fz, iy = (int)fy, ix = (int)fx;
    const int z0 = min(max(iz,     0), DIN - 1);
    const int z1 = min(max(iz + 1, 0), DIN - 1);
    const int y0 = min(max(iy,     0), DIN - 1);
    const int y1 = min(max(iy + 1, 0), DIN - 1);
    const int x0 = min(max(ix,     0), DIN - 1);
    const int x1 = min(max(ix + 1, 0), DIN - 1);

    // Element offset = (b<<24) | (z<<17) | (y<<10) | (x<<3); fits easily in i32.
    const int bofs = (p >= PER_BATCH) ? (1 << 24) : 0;
    const int oz0 = z0 << 17, oz1 = z1 << 17;
    const int oy0 = y0 << 10, oy1 = y1 << 10;
    const int ox0 = x0 << 3,  ox1 = x1 << 3;

    // f4 indices (element offset / 4); corner order = reference code 0..7:
    // code bits: (z,y,x) with z = bit2, y = bit1, x = bit0.
    int idx[8];
    idx[0] = (bofs + oz0 + oy0 + ox0) >> 2;
    idx[1] = (bofs + oz0 + oy0 + ox1) >> 2;
    idx[2] = (bofs + oz0 + oy1 + ox0) >> 2;
    idx[3] = (bofs + oz0 + oy1 + ox1) >> 2;
    idx[4] = (bofs + oz1 + oy0 + ox0) >> 2;
    idx[5] = (bofs + oz1 + oy0 + ox1) >> 2;
    idx[6] = (bofs + oz1 + oy1 + ox0) >> 2;
    idx[7] = (bofs + oz1 + oy1 + ox1) >> 2;

    // ---- Gather clause: 16 x global_load_b128, all in flight (L2-resident) ----
    const f4* __restrict__ in4 = (const f4*)in;
    f4 lo[8], hi[8];
#pragma unroll
    for (int k = 0; k < 8; ++k) {
        lo[k] = in4[idx[k]];        // channels 0..3
        hi[k] = in4[idx[k] + 1];    // channels 4..7
    }

    // ---- Weighted accumulation, same order/associativity as reference ----
    const float wzs[2] = {1.0f - wz, wz};
    const float wys[2] = {1.0f - wy, wy};
    const float wxs[2] = {1.0f - wx, wx};

    f4 a0 = {0.0f, 0.0f, 0.0f, 0.0f};
    f4 a1 = {0.0f, 0.0f, 0.0f, 0.0f};
#pragma unroll
    for (int code = 0; code < 8; ++code) {
        const float wt = wzs[(code >> 2) & 1] * wys[(code >> 1) & 1] * wxs[code & 1];
        a0 += lo[code] * wt;
        a1 += hi[code] * wt;
    }

    // ---- Streaming NT stores: 2 x global_store_b128 th:NT ----
    f4* __restrict__ out4 = (f4*)out;
    __builtin_nontemporal_store(a0, out4 + 2 * p);
    __builtin_nontemporal_store(a1, out4 + 2 * p + 1);
}

extern "C" void kernel_launch(void* const* d_in, const int* in_sizes, int n_in,
                              void* d_out, int out_size, void* d_ws, size_t ws_size,
                              hipStream_t stream) {
    const float* in     = (const float*)d_in[0];   // (2,128,128,128,8) f32
    const float* coords = (const float*)d_in[1];   // (2,96,96,96,3) f32
    float* out          = (float*)d_out;           // (2,96,96,96,8) f32

    const int nblocks = NPTS / THREADS;            // 6912, exact
    resample_trilinear_kernel<<<nblocks, THREADS, COORD_BYTES, stream>>>(in, coords, out);
}